// Attention_block_26645977105007
// MI455X (gfx1250) — compile-verified
//
#include <hip/hip_runtime.h>
#include <hip/hip_bf16.h>
#include <math.h>

// ---------------------------------------------------------------------------
// GATConv forward (PyG-style, with self loops + per-dst mean edge_attr loops)
// N=50000 D=128 H=4 C=16 (HC=64), E=800000 ED=64
// ---------------------------------------------------------------------------

typedef __attribute__((ext_vector_type(2))) float v2f;
typedef __attribute__((ext_vector_type(8))) float v8f;

#define HC 64
#define DIN 128
#define EDIM 64
#define NHEAD 4
#define NEG_SLOPE 0.2f

// ---- monotone float <-> uint encoding for atomicMax-based segment max -----
__device__ __forceinline__ unsigned fenc(float f) {
  unsigned u = __float_as_uint(f);
  return (u & 0x80000000u) ? ~u : (u | 0x80000000u);
}
__device__ __forceinline__ float fdec(unsigned u) {
  return (u & 0x80000000u) ? __uint_as_float(u & 0x7FFFFFFFu)
                           : __uint_as_float(~u);
}
__device__ __forceinline__ float lrelu(float a) {
  return a > 0.f ? a : NEG_SLOPE * a;
}

// ---------------------------------------------------------------------------
__global__ void gat_zero_ws(float* ws, size_t n) {
  size_t i = (size_t)blockIdx.x * blockDim.x + threadIdx.x;
  size_t stride = (size_t)gridDim.x * blockDim.x;
  for (; i < n; i += stride) ws[i] = 0.0f;
}

// v[h*64+d] = sum_c att_edge[h*16+c] * W_edge[(h*16+c)*64 + d]
__global__ void gat_fold_edge_att(const float* __restrict__ W_edge,
                                  const float* __restrict__ att_edge,
                                  float* __restrict__ v) {
  int t = threadIdx.x;           // 0..255 = (h, d)
  int h = t >> 6, d = t & 63;
  float s = 0.f;
#pragma unroll
  for (int c = 0; c < 16; ++c)
    s += att_edge[h * 16 + c] * W_edge[(size_t)(h * 16 + c) * EDIM + d];
  v[t] = s;
}

// scatter: loop_sum[dst] += edge_attr[e], cnt[dst] += 1
__global__ void gat_loop_gather(const float* __restrict__ ea,
                                const int* __restrict__ ei, int E,
                                float* __restrict__ loop_sum,
                                float* __restrict__ cnt) {
  long long t = (long long)blockIdx.x * blockDim.x + threadIdx.x;
  long long total = (long long)E * EDIM;
  if (t >= total) return;
  int j = (int)(t & 63);
  int e = (int)(t >> 6);
  int d = ei[E + e];
  atomicAdd(&loop_sum[(size_t)d * EDIM + j], ea[(size_t)e * EDIM + j]);
  if (j == 0) atomicAdd(&cnt[d], 1.0f);
}

// ---------------------------------------------------------------------------
// xh = x @ W^T via V_WMMA_F32_16X16X4_F32.  Each wave: 16-row x 64-col strip.
// A 16x4 fp32: lane = 16*half + r holds row M=r, K = 2*half..2*half+1.
// B  4x16 fp32: lane holds col N=r, rows K = 2*half..2*half+1.
// C/D 16x16:  acc[j] = D[j + 8*half][r].
// ---------------------------------------------------------------------------
__global__ __launch_bounds__(128) void gat_xh_wmma(
    const float* __restrict__ x, const float* __restrict__ W,
    float* __restrict__ xh, int n_rows) {
  const int wave = threadIdx.x >> 5;
  const int lane = threadIdx.x & 31;
  const int mtile = blockIdx.x * 4 + wave;
  const int mtiles = (n_rows + 15) >> 4;
  if (mtile >= mtiles) return;            // wave-uniform: EXEC stays all-1s
  const int m0 = mtile * 16;
  const int half = lane >> 4;
  const int r = lane & 15;

  int arow = m0 + r;
  if (arow >= n_rows) arow = n_rows - 1;  // clamp loads, keep EXEC full
  const float* xrow = x + (size_t)arow * DIN;
  const float* w0 = W + (size_t)(0 * 16 + r) * DIN;
  const float* w1 = W + (size_t)(1 * 16 + r) * DIN;
  const float* w2 = W + (size_t)(2 * 16 + r) * DIN;
  const float* w3 = W + (size_t)(3 * 16 + r) * DIN;

  v8f acc0 = {}, acc1 = {}, acc2 = {}, acc3 = {};
  for (int k = 0; k < DIN; k += 4) {
    const int kk = k + 2 * half;
    v2f a;  a.x = xrow[kk]; a.y = xrow[kk + 1];
    v2f b0; b0.x = w0[kk];  b0.y = w0[kk + 1];
    v2f b1; b1.x = w1[kk];  b1.y = w1[kk + 1];
    v2f b2; b2.x = w2[kk];  b2.y = w2[kk + 1];
    v2f b3; b3.x = w3[kk];  b3.y = w3[kk + 1];
    acc0 = __builtin_amdgcn_wmma_f32_16x16x4_f32(false, a, false, b0, (short)0, acc0, false, false);
    acc1 = __builtin_amdgcn_wmma_f32_16x16x4_f32(false, a, false, b1, (short)0, acc1, false, false);
    acc2 = __builtin_amdgcn_wmma_f32_16x16x4_f32(false, a, false, b2, (short)0, acc2, false, false);
    acc3 = __builtin_amdgcn_wmma_f32_16x16x4_f32(false, a, false, b3, (short)0, acc3, false, false);
  }
#pragma unroll
  for (int j = 0; j < 8; ++j) {
    int row = m0 + j + 8 * half;
    if (row < n_rows) {
      float* o = xh + (size_t)row * HC + r;
      o[0]  = acc0[j];
      o[16] = acc1[j];
      o[32] = acc2[j];
      o[48] = acc3[j];
    }
  }
}

// a_src[n,h] = sum_c xh[n,h,c]*att_src[h,c]  (same for a_dst)
__global__ void gat_node_att(const float* __restrict__ xh,
                             const float* __restrict__ att_src,
                             const float* __restrict__ att_dst,
                             float* __restrict__ asrc,
                             float* __restrict__ adst, int N) {
  int n = blockIdx.x * blockDim.x + threadIdx.x;
  if (n >= N) return;
  const float* row = xh + (size_t)n * HC;
#pragma unroll
  for (int h = 0; h < NHEAD; ++h) {
    float s = 0.f, d = 0.f;
#pragma unroll
    for (int c = 0; c < 16; ++c) {
      float xv = row[h * 16 + c];
      s += xv * att_src[h * 16 + c];
      d += xv * att_dst[h * 16 + c];
    }
    asrc[(size_t)n * NHEAD + h] = s;
    adst[(size_t)n * NHEAD + h] = d;
  }
}

// self-loop edges: alpha[E+n] from mean incoming edge_attr; amax update
__global__ void gat_self_alpha(const float* __restrict__ loop_sum,
                               const float* __restrict__ cnt,
                               const float* __restrict__ asrc,
                               const float* __restrict__ adst,
                               const float* __restrict__ v,
                               float* __restrict__ alpha,
                               unsigned* __restrict__ amax, int N, int E) {
  int n = blockIdx.x * blockDim.x + threadIdx.x;
  if (n >= N) return;
  float c = cnt[n];
  float inv = 1.0f / (c > 1.0f ? c : 1.0f);
  float ae[NHEAD] = {0.f, 0.f, 0.f, 0.f};
  const float* row = loop_sum + (size_t)n * EDIM;
  for (int j = 0; j < EDIM; ++j) {
    float la = row[j] * inv;
#pragma unroll
    for (int h = 0; h < NHEAD; ++h) ae[h] += la * v[h * EDIM + j];
  }
#pragma unroll
  for (int h = 0; h < NHEAD; ++h) {
    float a = lrelu(asrc[(size_t)n * NHEAD + h] +
                    adst[(size_t)n * NHEAD + h] + ae[h]);
    alpha[(size_t)(E + n) * NHEAD + h] = a;
    atomicMax(&amax[(size_t)n * NHEAD + h], fenc(a));
  }
}

// real edges: a_edge = ea . v, alpha = lrelu(asrc[s]+adst[d]+ae); amax update
__global__ void gat_edge_alpha(const float* __restrict__ ea,
                               const int* __restrict__ ei,
                               const float* __restrict__ asrc,
                               const float* __restrict__ adst,
                               const float* __restrict__ v,
                               float* __restrict__ alpha,
                               unsigned* __restrict__ amax, int E) {
  int e = blockIdx.x * blockDim.x + threadIdx.x;
  if (e >= E) return;
  int s = ei[e];
  int d = ei[E + e];
  const float4* row = reinterpret_cast<const float4*>(ea + (size_t)e * EDIM);
  float ae0 = 0.f, ae1 = 0.f, ae2 = 0.f, ae3 = 0.f;
#pragma unroll 4
  for (int q = 0; q < 16; ++q) {
    float4 x4 = row[q];
    int j = q * 4;
    ae0 += x4.x * v[0 * 64 + j] + x4.y * v[0 * 64 + j + 1] +
           x4.z * v[0 * 64 + j + 2] + x4.w * v[0 * 64 + j + 3];
    ae1 += x4.x * v[1 * 64 + j] + x4.y * v[1 * 64 + j + 1] +
           x4.z * v[1 * 64 + j + 2] + x4.w * v[1 * 64 + j + 3];
    ae2 += x4.x * v[2 * 64 + j] + x4.y * v[2 * 64 + j + 1] +
           x4.z * v[2 * 64 + j + 2] + x4.w * v[2 * 64 + j + 3];
    ae3 += x4.x * v[3 * 64 + j] + x4.y * v[3 * 64 + j + 1] +
           x4.z * v[3 * 64 + j + 2] + x4.w * v[3 * 64 + j + 3];
  }
  float aeh[NHEAD] = {ae0, ae1, ae2, ae3};
  float4 out;
  float* po = (float*)&out;
#pragma unroll
  for (int h = 0; h < NHEAD; ++h) {
    float a = lrelu(asrc[(size_t)s * NHEAD + h] +
                    adst[(size_t)d * NHEAD + h] + aeh[h]);
    po[h] = a;
    atomicMax(&amax[(size_t)d * NHEAD + h], fenc(a));
  }
  reinterpret_cast<float4*>(alpha)[e] = out;
}

// alpha <- exp(alpha - amax[dst]); denom[dst] += alpha
__global__ void gat_expsum(float* __restrict__ alpha,
                           const unsigned* __restrict__ amax,
                           float* __restrict__ denom,
                           const int* __restrict__ ei, int E, int N) {
  int e = blockIdx.x * blockDim.x + threadIdx.x;
  if (e >= E + N) return;
  int d = (e < E) ? ei[E + e] : (e - E);
  float4 a4 = reinterpret_cast<const float4*>(alpha)[e];
  float* pa = (float*)&a4;
#pragma unroll
  for (int h = 0; h < NHEAD; ++h) {
    float m = fdec(amax[(size_t)d * NHEAD + h]);
    float p = __expf(pa[h] - m);
    pa[h] = p;
    atomicAdd(&denom[(size_t)d * NHEAD + h], p);
  }
  reinterpret_cast<float4*>(alpha)[e] = a4;
}

// out_acc[dst, j] += xh[src, j] * alpha[e,h] / (denom[dst,h]+eps)
__global__ void gat_scatter(const float* __restrict__ xh,
                            const float* __restrict__ alpha,
                            const float* __restrict__ denom,
                            const int* __restrict__ ei, int E, int N,
                            float* __restrict__ oacc) {
  long long t = (long long)blockIdx.x * blockDim.x + threadIdx.x;
  long long total = (long long)(E + N) * HC;
  if (t >= total) return;
  int j = (int)(t & 63);
  int e = (int)(t >> 6);
  int h = j >> 4;
  int s, d;
  if (e < E) { s = ei[e]; d = ei[E + e]; } else { s = d = e - E; }
  float w = alpha[(size_t)e * NHEAD + h] /
            (denom[(size_t)d * NHEAD + h] + 1e-16f);
  atomicAdd(&oacc[(size_t)d * HC + j], xh[(size_t)s * HC + j] * w);
}

__global__ void gat_finalize(const float* __restrict__ oacc,
                             const float* __restrict__ bias,
                             float* __restrict__ out, int N) {
  long long t = (long long)blockIdx.x * blockDim.x + threadIdx.x;
  long long total = (long long)N * HC;
  if (t >= total) return;
  out[t] = oacc[t] + bias[(int)(t & 63)];
}

// ---------------------------------------------------------------------------
extern "C" void kernel_launch(void* const* d_in, const int* in_sizes, int n_in,
                              void* d_out, int out_size, void* d_ws,
                              size_t ws_size, hipStream_t stream) {
  const float* x        = (const float*)d_in[0];
  const int*   ei       = (const int*)d_in[1];   // [2,E] (src row, dst row)
  const float* ea       = (const float*)d_in[2];
  const float* W        = (const float*)d_in[3];
  const float* W_edge   = (const float*)d_in[4];
  const float* att_src  = (const float*)d_in[5];
  const float* att_dst  = (const float*)d_in[6];
  const float* att_edge = (const float*)d_in[7];
  const float* bias     = (const float*)d_in[8];
  float* out = (float*)d_out;

  const int N = in_sizes[0] / DIN;
  const int E = in_sizes[1] / 2;

  // workspace layout (floats)
  float* ws = (float*)d_ws;
  size_t o = 0;
  float*    cnt   = ws + o; o += (size_t)N;
  float*    loop  = ws + o; o += (size_t)N * EDIM;
  float*    xh    = ws + o; o += (size_t)N * HC;
  float*    asrc  = ws + o; o += (size_t)N * NHEAD;
  float*    adst  = ws + o; o += (size_t)N * NHEAD;
  float*    v     = ws + o; o += (size_t)NHEAD * EDIM;
  float*    alpha = ws + o; o += (size_t)(E + N) * NHEAD;
  unsigned* amax  = (unsigned*)(ws + o); o += (size_t)N * NHEAD;
  float*    denom = ws + o; o += (size_t)N * NHEAD;
  float*    oacc  = ws + o; o += (size_t)N * HC;
  const size_t ws_used = o;

  const int B = 256;

  // 1) zero accumulators (amax = 0 == encoded -NaN floor, below all reals)
  gat_zero_ws<<<4096, B, 0, stream>>>(ws, ws_used);

  // 2) fold att_edge into W_edge: v[h,d]
  gat_fold_edge_att<<<1, 256, 0, stream>>>(W_edge, att_edge, v);

  // 3) per-dst counts + edge_attr sums (for self-loop mean attr)
  {
    long long tot = (long long)E * EDIM;
    gat_loop_gather<<<(unsigned)((tot + B - 1) / B), B, 0, stream>>>(
        ea, ei, E, loop, cnt);
  }

  // 4) xh = x @ W^T  (WMMA fp32, 16x16x4)
  {
    int mtiles = (N + 15) / 16;
    int blocks = (mtiles + 3) / 4;   // 4 waves/block, 1 strip/wave
    gat_xh_wmma<<<blocks, 128, 0, stream>>>(x, W, xh, N);
  }

  // 5) a_src / a_dst per node
  gat_node_att<<<(N + B - 1) / B, B, 0, stream>>>(xh, att_src, att_dst,
                                                  asrc, adst, N);

  // 6) alpha for self-loop edges + segment max
  gat_self_alpha<<<(N + B - 1) / B, B, 0, stream>>>(loop, cnt, asrc, adst, v,
                                                    alpha, amax, N, E);

  // 7) alpha for real edges + segment max
  gat_edge_alpha<<<(E + B - 1) / B, B, 0, stream>>>(ea, ei, asrc, adst, v,
                                                    alpha, amax, E);

  // 8) exp + segment sum
  gat_expsum<<<(E + N + B - 1) / B, B, 0, stream>>>(alpha, amax, denom, ei,
                                                    E, N);

  // 9) weighted scatter-add of messages
  {
    long long tot = (long long)(E + N) * HC;
    gat_scatter<<<(unsigned)((tot + B - 1) / B), B, 0, stream>>>(
        xh, alpha, denom, ei, E, N, oacc);
  }

  // 10) + bias -> out
  {
    long long tot = (long long)N * HC;
    gat_finalize<<<(unsigned)((tot + B - 1) / B), B, 0, stream>>>(oacc, bias,
                                                                  out, N);
  }
}